// KDTreeRouter_28750511079538
// MI455X (gfx1250) — compile-verified
//
#include <hip/hip_runtime.h>
#include <math.h>

typedef __attribute__((ext_vector_type(2))) float v2f;
typedef __attribute__((ext_vector_type(8))) float v8f;

#define N_TOK 16384
#define HDIM  4096
#define NEXP  64
#define PDIM  6
#define TOPK  8
#define WAVES 4
#define NCH   (HDIM / 64)
#define CSTR  (HDIM + 4)   /* padded comp row stride: 4100 % 64 == 4 -> conflict free */

// ---------------------------------------------------------------------------
// Prep kernel: mean_dot[p] = dot(pca_mean, comp[p]) (padded to 16), c2[e]=|cent_e|^2
// ws layout (floats): [0..15] mean_dot (p>=6 zero), [16..79] c2
// ---------------------------------------------------------------------------
__global__ void prep_kernel(const float* __restrict__ mean,
                            const float* __restrict__ comp,
                            const float* __restrict__ cent,
                            float* __restrict__ ws) {
  __shared__ float red[PDIM][256];
  const int t = threadIdx.x;
  float acc[PDIM];
#pragma unroll
  for (int p = 0; p < PDIM; ++p) acc[p] = 0.f;
  for (int h = t; h < HDIM; h += 256) {
    float m = mean[h];
#pragma unroll
    for (int p = 0; p < PDIM; ++p) acc[p] += m * comp[p * HDIM + h];
  }
#pragma unroll
  for (int p = 0; p < PDIM; ++p) red[p][t] = acc[p];
  __syncthreads();
  if (t < PDIM) {
    float s = 0.f;
    for (int i = 0; i < 256; ++i) s += red[t][i];
    ws[t] = s;
  } else if (t < 16) {
    ws[t] = 0.f;
  }
  if (t < NEXP) {
    float s = 0.f;
#pragma unroll
    for (int p = 0; p < PDIM; ++p) { float v = cent[t * PDIM + p]; s += v * v; }
    ws[16 + t] = s;
  }
}

// ---------------------------------------------------------------------------
// WMMA micro-kernel for one 16x64 LDS tile: fragment loads are software-
// pipelined one 2-WMMA group ahead so ds latency overlaps WMMA issue.
// ---------------------------------------------------------------------------
__device__ __forceinline__ v8f tile_wmma(const float* arow, const float* crow,
                                         v8f c) {
  v2f a0, a1, b0, b1, na0, na1, nb0, nb1;
  a0 = *(const v2f*)(arow + 0);
  a1 = *(const v2f*)(arow + 4);
  b0 = *(const v2f*)(crow + 0);
  b1 = *(const v2f*)(crow + 4);
#pragma unroll
  for (int g = 0; g < 8; ++g) {           // 8 groups x 2 WMMAs = K of 64
    if (g < 7) {
      na0 = *(const v2f*)(arow + (2 * g + 2) * 4);
      na1 = *(const v2f*)(arow + (2 * g + 3) * 4);
      nb0 = *(const v2f*)(crow + (2 * g + 2) * 4);
      nb1 = *(const v2f*)(crow + (2 * g + 3) * 4);
    }
    c = __builtin_amdgcn_wmma_f32_16x16x4_f32(false, a0, false, b0, (short)0,
                                              c, false, false);
    c = __builtin_amdgcn_wmma_f32_16x16x4_f32(false, a1, false, b1, (short)0,
                                              c, false, false);
    a0 = na0; a1 = na1; b0 = nb0; b1 = nb1;
  }
  return c;
}

// ---------------------------------------------------------------------------
// Fused router: per-wave 16-token tile, f32 WMMA projection over H (explicit
// 2-stage global->LDS software pipeline), then cdist -> softmax -> top-k.
// out layout: [N*K] indices(float), [N*K] topk_probs, [N*E] probs
// ---------------------------------------------------------------------------
__global__ __launch_bounds__(WAVES * 32) void route_kernel(
    const float* __restrict__ x,
    const float* __restrict__ comp,
    const float* __restrict__ cent,
    const float* __restrict__ ws,
    float* __restrict__ out) {
  extern __shared__ float smem[];
  float* sComp = smem;                          // [PDIM][CSTR]
  float* sX    = sComp + PDIM * CSTR;           // [WAVES][2][16][68]
  float* sProj = sX + WAVES * 2 * 16 * 68;      // [WAVES][16][17]
  float* sProb = sProj + WAVES * 16 * 17;       // [WAVES][16][65]
  float* sCent = sProb + WAVES * 16 * 65;       // [NEXP][8]  (p0..p5, |c|^2, pad)
  float* sMean = sCent + NEXP * 8;              // [16]
  float* sZero = sMean + 16;                    // [68] zero pad for B lanes n>=6

  const int tid  = threadIdx.x;
  const int wave = tid >> 5;
  const int lane = tid & 31;
  const int lo   = lane & 15;
  const int hi   = lane >> 4;

  // Preload PCA components into padded LDS (coalesced, once per block).
  for (int i = tid; i < PDIM * HDIM; i += WAVES * 32) {
    int p = i >> 12;            // i / HDIM
    int h = i & (HDIM - 1);
    sComp[p * CSTR + h] = comp[i];
  }
  if (tid < NEXP) {
#pragma unroll
    for (int p = 0; p < PDIM; ++p) sCent[tid * 8 + p] = cent[tid * PDIM + p];
    sCent[tid * 8 + 6] = ws[16 + tid];
    sCent[tid * 8 + 7] = 0.f;
  }
  if (tid < 16) sMean[tid] = ws[tid];
  if (tid < 68) sZero[tid] = 0.f;
  __syncthreads();

  const int tokenBase = (blockIdx.x * WAVES + wave) * 16;
  const float* xbase  = x + (size_t)tokenBase * HDIM;
  float* tile0        = sX + wave * (2 * 16 * 68);
  float* tile1        = tile0 + 16 * 68;

  // B-fragment source: column n = lo of B (== PCA row lo). Lanes with lo>=6
  // read a zeroed LDS pad (broadcast, stride 0) -> no masking in the hot loop.
  const float* bbase = (lo < PDIM) ? (sComp + (size_t)lo * CSTR + hi * 2)
                                   : (sZero + hi * 2);
  const int    bstep = (lo < PDIM) ? 64 : 0;

  v8f c = {};

  float4 vA[8], vB[8];
  // Prologue: load chunk 0 into vA.
#pragma unroll
  for (int i = 0; i < 8; ++i) {
    int f = i * 32 + lane;
    vA[i] = *(const float4*)(xbase + (size_t)(f >> 4) * HDIM + ((f & 15) << 2));
  }

#pragma unroll 1
  for (int ch = 0; ch < NCH; ch += 2) {
    // ---- stage A: commit chunk ch, prefetch ch+1, WMMA on ch ----
#pragma unroll
    for (int i = 0; i < 8; ++i) {
      int f = i * 32 + lane;
      *(float4*)(tile0 + (f >> 4) * 68 + ((f & 15) << 2)) = vA[i];
    }
    {
      const float* src = xbase + (ch + 1) * 64;
#pragma unroll
      for (int i = 0; i < 8; ++i) {
        int f = i * 32 + lane;
        vB[i] = *(const float4*)(src + (size_t)(f >> 4) * HDIM + ((f & 15) << 2));
      }
    }
    c = tile_wmma(tile0 + lo * 68 + hi * 2, bbase + ch * bstep, c);

    // ---- stage B: commit chunk ch+1, prefetch ch+2 (clamped), WMMA on ch+1 ----
#pragma unroll
    for (int i = 0; i < 8; ++i) {
      int f = i * 32 + lane;
      *(float4*)(tile1 + (f >> 4) * 68 + ((f & 15) << 2)) = vB[i];
    }
    {
      int nc = (ch + 2 < NCH) ? (ch + 2) : (NCH - 1);   // clamped harmless reload
      const float* src = xbase + nc * 64;
#pragma unroll
      for (int i = 0; i < 8; ++i) {
        int f = i * 32 + lane;
        vA[i] = *(const float4*)(src + (size_t)(f >> 4) * HDIM + ((f & 15) << 2));
      }
    }
    c = tile_wmma(tile1 + lo * 68 + hi * 2, bbase + (ch + 1) * bstep, c);
  }

  // D layout: lane holds D[m = 8*hi + j][n = lo] in c[j]. Apply -mean_dot, stash.
  float* projW = sProj + wave * (16 * 17);
  {
    float md = sMean[lo];
#pragma unroll
    for (int j = 0; j < 8; ++j) projW[(hi * 8 + j) * 17 + lo] = c[j] - md;
  }

  // Epilogue: one lane per token (same wave; LDS ops are in-order, no barrier).
  if (lane < 16) {
    const int token = tokenBase + lane;
    float pr[PDIM];
    float s2 = 0.f;
#pragma unroll
    for (int p = 0; p < PDIM; ++p) {
      pr[p] = projW[lane * 17 + p];
      s2 += pr[p] * pr[p];
    }
    float* pb = sProb + wave * (16 * 65) + lane * 65;
    float mind = 3.4e38f;
    for (int e = 0; e < NEXP; ++e) {
      float dot = 0.f;
#pragma unroll
      for (int p = 0; p < PDIM; ++p) dot += pr[p] * sCent[e * 8 + p];
      float d2 = s2 - 2.f * dot + sCent[e * 8 + 6];
      float d = sqrtf(fmaxf(d2, 0.f));
      pb[e] = d;
      mind = fminf(mind, d);
    }
    float sum = 0.f;
    for (int e = 0; e < NEXP; ++e) {
      float pe = __expf(mind - pb[e]);     // stable softmax of -dist
      pb[e] = pe;
      sum += pe;
    }
    float inv = 1.f / sum;
    float* outP = out + 2 * (size_t)N_TOK * TOPK + (size_t)token * NEXP;
    for (int e = 0; e < NEXP; ++e) {
      float pe = pb[e] * inv;
      pb[e] = pe;
      outP[e] = pe;
    }
    // top-k: 8 stable max scans (strict '>' keeps lowest index on ties)
    float tv[TOPK]; int ti[TOPK];
    float acc = 0.f;
#pragma unroll
    for (int k = 0; k < TOPK; ++k) {
      float bv = -1.f; int bi = 0;
      for (int e = 0; e < NEXP; ++e) {
        float vv = pb[e];
        if (vv > bv) { bv = vv; bi = e; }
      }
      pb[bi] = -2.f;
      tv[k] = bv; ti[k] = bi; acc += bv;
    }
    float rn = 1.f / acc;
    float* outI = out + (size_t)token * TOPK;
    float* outV = out + (size_t)N_TOK * TOPK + (size_t)token * TOPK;
#pragma unroll
    for (int k = 0; k < TOPK; ++k) {
      outI[k] = (float)ti[k];
      outV[k] = tv[k] * rn;
    }
  }
}

// ---------------------------------------------------------------------------
extern "C" void kernel_launch(void* const* d_in, const int* in_sizes, int n_in,
                              void* d_out, int out_size, void* d_ws, size_t ws_size,
                              hipStream_t stream) {
  const float* x    = (const float*)d_in[0];
  const float* mean = (const float*)d_in[1];
  const float* comp = (const float*)d_in[2];
  const float* cent = (const float*)d_in[3];
  (void)in_sizes; (void)n_in; (void)out_size; (void)ws_size;
  float* ws  = (float*)d_ws;
  float* out = (float*)d_out;

  prep_kernel<<<1, 256, 0, stream>>>(mean, comp, cent, ws);

  const size_t ldsBytes =
      (size_t)(PDIM * CSTR + WAVES * 2 * 16 * 68 + WAVES * 16 * 17 +
               WAVES * 16 * 65 + NEXP * 8 + 16 + 68) * sizeof(float);
  route_kernel<<<N_TOK / (WAVES * 16), WAVES * 32, ldsBytes, stream>>>(
      x, comp, cent, ws, out);
}